// SimpleGCN_59768764891876
// MI455X (gfx1250) — compile-verified
//
#include <hip/hip_runtime.h>
#include <hip/hip_bf16.h>

#define N_NODES 100000
#define N_EDGES 3200000
#define F_IN 128
#define HID 32
#define N_GRAPHS 64

typedef __attribute__((ext_vector_type(2))) float v2f;
typedef __attribute__((ext_vector_type(8))) float v8f;

// ---------------------------------------------------------------------------
// Degree count (targets get +1 per incoming edge); self-loop handled in dinv.
// ---------------------------------------------------------------------------
__global__ void gcn_degree_kernel(const int* __restrict__ col,
                                  float* __restrict__ deg) {
    int e = blockIdx.x * blockDim.x + threadIdx.x;
    if (e < N_EDGES) atomicAdd(&deg[col[e]], 1.0f);
}

__global__ void gcn_dinv_kernel(float* __restrict__ deg) {
    int n = blockIdx.x * blockDim.x + threadIdx.x;
    if (n < N_NODES) deg[n] = rsqrtf(deg[n] + 1.0f);  // +1 self loop
}

// ---------------------------------------------------------------------------
// Dense transform C[M,32] = A[M,K] @ W[K,32] via V_WMMA_F32_16X16X4_F32.
// One wave computes one 16x16 C tile (two waves cover the 32-wide output).
// A fragment (16x4 f32, 2 VGPRs): lanes 0-15 -> K=k0,k0+1; lanes 16-31 -> K=k0+2,k0+3
// B fragment (4x16 f32, 2 VGPRs): mirrored K split, N = lane&15
// C (8 VGPRs): row = tile*16 + r + 8*(lane>>4), col = n0 + (lane&15)
// ---------------------------------------------------------------------------
__global__ void gcn_gemm_wmma_f32(const float* __restrict__ A,
                                  const float* __restrict__ W,
                                  float* __restrict__ C,
                                  int M, int K) {
    const int lane  = threadIdx.x & 31;
    const int wave  = threadIdx.x >> 5;                 // 0..7 (8 waves/block)
    const int mTile = blockIdx.x * 4 + (wave >> 1);     // 4 M-tiles per block
    const int n0    = (wave & 1) * 16;                  // N half
    if (mTile * 16 >= M) return;                        // wave-uniform branch

    const int aRow  = mTile * 16 + (lane & 15);
    const int kHalf = (lane >> 4) * 2;                  // 0 or 2
    const int bCol  = n0 + (lane & 15);

    v8f acc = {};
    for (int k0 = 0; k0 < K; k0 += 4) {
        v2f a, b;
        const float* ap = A + (size_t)aRow * K + (k0 + kHalf);
        a.x = ap[0];
        a.y = ap[1];
        const float* bp = W + (size_t)(k0 + kHalf) * HID + bCol;
        b.x = bp[0];
        b.y = bp[HID];
        acc = __builtin_amdgcn_wmma_f32_16x16x4_f32(
            /*neg_a=*/false, a, /*neg_b=*/false, b,
            /*c_mod=*/(short)0, acc, /*reuse_a=*/false, /*reuse_b=*/false);
    }

    const int mBase = mTile * 16 + ((lane >> 4) * 8);
#pragma unroll
    for (int r = 0; r < 8; ++r) {
        C[(size_t)(mBase + r) * HID + bCol] = acc[r];
    }
}

// ---------------------------------------------------------------------------
// Edge scatter: one wave per edge, one lane per feature (HID == 32 == wave32).
// agg[col] += H[row] * dinv[row]*dinv[col]. Accumulator fits in 192MB L2.
// ---------------------------------------------------------------------------
__global__ void gcn_edge_agg_kernel(const int* __restrict__ row,
                                    const int* __restrict__ col,
                                    const float* __restrict__ H,
                                    const float* __restrict__ dinv,
                                    float* __restrict__ agg) {
    int gid = blockIdx.x * blockDim.x + threadIdx.x;
    int e = gid >> 5;
    int f = gid & 31;
    if (e >= N_EDGES) return;
    int r = row[e];
    int c = col[e];
    float coef = dinv[r] * dinv[c];
    atomicAdd(&agg[(size_t)c * HID + f], H[(size_t)r * HID + f] * coef);
}

// ---------------------------------------------------------------------------
// Layer 1 epilogue: agg += H * dinv^2 (self loop) + bias, then ReLU in place.
// ---------------------------------------------------------------------------
__global__ void gcn_selfloop_relu_kernel(const float* __restrict__ H,
                                         const float* __restrict__ dinv,
                                         const float* __restrict__ bias,
                                         float* __restrict__ agg) {
    int i = blockIdx.x * blockDim.x + threadIdx.x;
    if (i >= N_NODES * HID) return;
    int n = i >> 5;
    int f = i & 31;
    float d = dinv[n];
    float v = agg[i] + H[i] * d * d + bias[f];
    agg[i] = fmaxf(v, 0.0f);
}

// ---------------------------------------------------------------------------
// Layer 2 epilogue: same as above, but also accumulates the per-graph mean
// pool (sum + count) with atomics into tiny [64,32] buffers.
// ---------------------------------------------------------------------------
__global__ void gcn_final_node_kernel(const float* __restrict__ H,
                                      const float* __restrict__ dinv,
                                      const float* __restrict__ bias,
                                      const int* __restrict__ batch,
                                      float* __restrict__ agg,     // d_out nodes
                                      float* __restrict__ pooled,  // [64,32]
                                      float* __restrict__ cnt) {   // [64]
    int i = blockIdx.x * blockDim.x + threadIdx.x;
    if (i >= N_NODES * HID) return;
    int n = i >> 5;
    int f = i & 31;
    float d = dinv[n];
    float v = agg[i] + H[i] * d * d + bias[f];
    v = fmaxf(v, 0.0f);
    agg[i] = v;
    int g = batch[n];
    atomicAdd(&pooled[g * HID + f], v);
    if (f == 0) atomicAdd(&cnt[g], 1.0f);
}

// ---------------------------------------------------------------------------
// Graph head: sigmoid(mean_pool @ Wc + bc), 64 graphs, 1 thread each.
// ---------------------------------------------------------------------------
__global__ void gcn_graph_pred_kernel(const float* __restrict__ pooled,
                                      const float* __restrict__ cnt,
                                      const float* __restrict__ Wc,
                                      const float* __restrict__ bc,
                                      float* __restrict__ out) {
    int g = blockIdx.x * blockDim.x + threadIdx.x;
    if (g >= N_GRAPHS) return;
    float inv = 1.0f / fmaxf(cnt[g], 1.0f);
    float s = bc[0];
#pragma unroll
    for (int f = 0; f < HID; ++f) {
        s += pooled[g * HID + f] * inv * Wc[f];
    }
    out[g] = 1.0f / (1.0f + __expf(-s));
}

extern "C" void kernel_launch(void* const* d_in, const int* in_sizes, int n_in,
                              void* d_out, int out_size, void* d_ws, size_t ws_size,
                              hipStream_t stream) {
    const float* x     = (const float*)d_in[0];
    const int*   eidx  = (const int*)d_in[1];
    const int*   row   = eidx;             // edge_index[0]
    const int*   col   = eidx + N_EDGES;   // edge_index[1]
    const int*   batch = (const int*)d_in[2];
    const float* W1    = (const float*)d_in[3];
    const float* b1    = (const float*)d_in[4];
    const float* W2    = (const float*)d_in[5];
    const float* b2    = (const float*)d_in[6];
    const float* Wc    = (const float*)d_in[7];
    const float* bc    = (const float*)d_in[8];
    float*       out   = (float*)d_out;

    // Workspace layout (floats): dinv | H (N*32) | agg1/h (N*32) | pooled | cnt
    float* ws     = (float*)d_ws;
    float* dinv   = ws;
    float* Hbuf   = dinv + N_NODES;
    float* Abuf   = Hbuf + (size_t)N_NODES * HID;
    float* pooled = Abuf + (size_t)N_NODES * HID;
    float* cnt    = pooled + N_GRAPHS * HID;

    const int mTiles     = N_NODES / 16;            // 6250, exact
    const int gemmBlocks = (mTiles + 3) / 4;        // 4 M-tiles x 2 N-halves = 8 waves
    const int edgeGrid   = (N_EDGES * 32 + 255) / 256;
    const int nodeGrid   = (N_NODES * HID + 255) / 256;

    // --- normalization coefficients ---
    hipMemsetAsync(dinv, 0, N_NODES * sizeof(float), stream);
    gcn_degree_kernel<<<(N_EDGES + 255) / 256, 256, 0, stream>>>(col, dinv);
    gcn_dinv_kernel<<<(N_NODES + 255) / 256, 256, 0, stream>>>(dinv);

    // --- layer 1: H1 = x @ W1 (WMMA), aggregate, self-loop + bias + ReLU ---
    gcn_gemm_wmma_f32<<<gemmBlocks, 256, 0, stream>>>(x, W1, Hbuf, N_NODES, F_IN);
    hipMemsetAsync(Abuf, 0, (size_t)N_NODES * HID * sizeof(float), stream);
    gcn_edge_agg_kernel<<<edgeGrid, 256, 0, stream>>>(row, col, Hbuf, dinv, Abuf);
    gcn_selfloop_relu_kernel<<<nodeGrid, 256, 0, stream>>>(Hbuf, dinv, b1, Abuf);

    // --- layer 2: H2 = h @ W2 (WMMA, reuse Hbuf), aggregate into d_out ---
    gcn_gemm_wmma_f32<<<gemmBlocks, 256, 0, stream>>>(Abuf, W2, Hbuf, N_NODES, HID);
    hipMemsetAsync(out, 0, (size_t)N_NODES * HID * sizeof(float), stream);
    hipMemsetAsync(pooled, 0, (N_GRAPHS * HID + N_GRAPHS) * sizeof(float), stream);
    gcn_edge_agg_kernel<<<edgeGrid, 256, 0, stream>>>(row, col, Hbuf, dinv, out);
    gcn_final_node_kernel<<<nodeGrid, 256, 0, stream>>>(Hbuf, dinv, b2, batch,
                                                        out, pooled, cnt);

    // --- graph head ---
    gcn_graph_pred_kernel<<<1, N_GRAPHS, 0, stream>>>(pooled, cnt, Wc, bc,
                                                      out + (size_t)N_NODES * HID);
}